// EMA_layer_2585570312709
// MI455X (gfx1250) — compile-verified
//
#include <hip/hip_runtime.h>
#include <math.h>

// Problem constants (from the reference): x[T,D]; h_a,h_s,alpha [N,D]
#define T_DIM   4096
#define N_DIM   16
#define D_DIM   512
#define NC      (N_DIM * D_DIM)        // 8192 chains (n,d)
#define ROW2    (2 * NC)               // one output row [2N, D] = 16384 floats
#define R_ROWS  8                      // timesteps per async-staged LDS slab
#define NB_N    4                      // n-values sharing one block / x-slab

// Native clang vector type: required for __builtin_nontemporal_store
// (HIP's float4 is a struct and is rejected by the builtin).
typedef __attribute__((ext_vector_type(4))) float v4f;

// ---------------------------------------------------------------------------
// Phase 1: per-chunk affine/quadratic transfer coefficients.
// Within a chunk of length L starting from unknown (ha0, hv0):
//   ha_t = a_t*ha0 + b_t          (a_t = (1-al)^t)
//   hv_t = a_t*hv0 + q0 + q1*ha0 + q2*ha0^2
// Accumulated sequentially over the chunk; all chunks in parallel.
// ws layout (floats): A[C*NC] B[C*NC] Q0[C*NC] Q1[C*NC] Q2[C*NC] Sha[C*NC] Shv[C*NC]
// ---------------------------------------------------------------------------
__global__ void __launch_bounds__(512)
ema_phase1(const float* __restrict__ X, const float* __restrict__ alpha,
           float* __restrict__ ws, int C, int L) {
  const int tid = threadIdx.x;
  const int dq  = tid & 127;          // d-quad index
  const int nl  = tid >> 7;           // 0..3
  const int g   = blockIdx.x;         // grid = C * (N/NB_N) = C*4
  const int c   = g >> 2;
  const int n   = (g & 3) * NB_N + nl;
  const int d   = dq * 4;
  const int chain = n * D_DIM + d;

  float4 av = *(const float4*)(alpha + chain);
  float al[4] = {av.x, av.y, av.z, av.w};
  float om[4], a[4], b[4], q0[4], q1[4], q2[4];
#pragma unroll
  for (int j = 0; j < 4; ++j) {
    om[j] = 1.0f - al[j];
    a[j] = 1.0f; b[j] = 0.0f; q0[j] = 0.0f; q1[j] = 0.0f; q2[j] = 0.0f;
  }

  const float* xrow = X + (size_t)c * (size_t)L * D_DIM + d;
  for (int t = 0; t < L; ++t) {
    float4 xv = *(const float4*)xrow;
    xrow += D_DIM;
    float xx[4] = {xv.x, xv.y, xv.z, xv.w};
#pragma unroll
    for (int j = 0; j < 4; ++j) {
      float u = xx[j] - b[j];                          // d_t = u - a*ha0
      q0[j] = om[j] * fmaf(al[j] * u, u, q0[j]);
      q1[j] = om[j] * fmaf(-2.0f * al[j] * u, a[j], q1[j]);
      q2[j] = om[j] * fmaf(al[j] * a[j], a[j], q2[j]);
      b[j]  = fmaf(al[j], xx[j], om[j] * b[j]);        // ha = al*x + (1-al)*ha
      a[j] *= om[j];
    }
  }

  const size_t CN   = (size_t)C * NC;
  const size_t base = (size_t)c * NC + chain;
  *(float4*)(ws + 0 * CN + base) = make_float4(a[0],  a[1],  a[2],  a[3]);
  *(float4*)(ws + 1 * CN + base) = make_float4(b[0],  b[1],  b[2],  b[3]);
  *(float4*)(ws + 2 * CN + base) = make_float4(q0[0], q0[1], q0[2], q0[3]);
  *(float4*)(ws + 3 * CN + base) = make_float4(q1[0], q1[1], q1[2], q1[3]);
  *(float4*)(ws + 4 * CN + base) = make_float4(q2[0], q2[1], q2[2], q2[3]);
}

// ---------------------------------------------------------------------------
// Phase 2: compose transfers across chunks -> per-chunk start states; tails.
// ---------------------------------------------------------------------------
__global__ void __launch_bounds__(256)
ema_phase2(const float* __restrict__ h_a, const float* __restrict__ h_s,
           float* __restrict__ ws, float* __restrict__ out, int C) {
  const int chain = blockIdx.x * blockDim.x + threadIdx.x;
  if (chain >= NC) return;
  const size_t CN = (size_t)C * NC;
  const float* A  = ws;
  const float* B  = ws + 1 * CN;
  const float* Q0 = ws + 2 * CN;
  const float* Q1 = ws + 3 * CN;
  const float* Q2 = ws + 4 * CN;
  float* Sha = ws + 5 * CN;
  float* Shv = ws + 6 * CN;

  float ha = h_a[chain];
  float s  = h_s[chain];
  float hv = s * s;
  for (int c = 0; c < C; ++c) {
    const size_t i = (size_t)c * NC + chain;
    Sha[i] = ha;
    Shv[i] = hv;
    const float Ac = A[i];
    const float hvn = fmaf(Ac, hv, fmaf(Q2[i], ha * ha, fmaf(Q1[i], ha, Q0[i])));
    ha = fmaf(Ac, ha, B[i]);
    hv = hvn;
  }
  const size_t tail = (size_t)T_DIM * (size_t)ROW2;
  out[tail + chain]      = ha;
  out[tail + NC + chain] = sqrtf(hv);
}

// ---------------------------------------------------------------------------
// Phase 3: replay the exact reference recurrence per chunk, streaming output.
// x slabs are staged into LDS with double-buffered async global->LDS copies
// (CDNA5 GLOBAL_LOAD_ASYNC_TO_LDS_B128 + s_wait_asynccnt), shared by NB_N=4
// n-values per block. Output uses non-temporal (streaming) stores: 256 MB of
// write-once data must not evict the reused x/scratch working set from L2.
// ---------------------------------------------------------------------------
__global__ void __launch_bounds__(512)
ema_phase3(const float* __restrict__ X, const float* __restrict__ alpha,
           const float* __restrict__ ws, float* __restrict__ out, int C, int L) {
  __shared__ __align__(16) float xs[2 * R_ROWS * D_DIM];   // 32 KB, double buffer

  const int tid = threadIdx.x;
  const int dq  = tid & 127;
  const int nl  = tid >> 7;
  const int g   = blockIdx.x;
  const int c   = g >> 2;
  const int n   = (g & 3) * NB_N + nl;
  const int d   = dq * 4;
  const int chain = n * D_DIM + d;

  const size_t CN = (size_t)C * NC;
  const float* Sha = ws + 5 * CN;
  const float* Shv = ws + 6 * CN;

  float4 av = *(const float4*)(alpha + chain);
  float al[4] = {av.x, av.y, av.z, av.w};
  float om[4];
#pragma unroll
  for (int j = 0; j < 4; ++j) om[j] = 1.0f - al[j];

  const size_t sbase = (size_t)c * NC + chain;
  float4 hav = *(const float4*)(Sha + sbase);
  float4 hvv = *(const float4*)(Shv + sbase);
  float ha[4] = {hav.x, hav.y, hav.z, hav.w};
  float hv[4] = {hvv.x, hvv.y, hvv.z, hvv.w};

  const float* xchunk = X + (size_t)c * (size_t)L * D_DIM;
  const bool stager = (tid < 128);                       // waves 0..3 issue copies
  const unsigned ldsBase = (unsigned)(uintptr_t)&xs[0];  // low 32 bits == LDS byte addr

  // One async b128 per lane per row (128 lanes x 16B = one 512-float row).
  // The instruction's IOFFSET is added to BOTH the global and LDS addresses
  // (CDNA5 ISA) and the row stride (2048B) is identical on
  // both sides, so all 8 rows share one base VGPR pair + immediate offsets.
  auto stage = [&](int s) {
    unsigned long long gp =
        (unsigned long long)(uintptr_t)(xchunk + (size_t)(s * R_ROWS) * D_DIM + d);
    unsigned lp = ldsBase + (unsigned)(((s & 1) * R_ROWS * D_DIM + d) * 4);
#pragma unroll
    for (int r = 0; r < R_ROWS; ++r) {
      asm volatile("global_load_async_to_lds_b128 %0, %1, off offset:%2"
                   :: "v"(lp), "v"(gp), "n"(r * D_DIM * 4) : "memory");
    }
  };

  if (stager) stage(0);
  const int S = L / R_ROWS;
  for (int s = 0; s < S; ++s) {
    if (stager) {
      if (s + 1 < S) {
        stage(s + 1);                                   // prefetch next slab
        asm volatile("s_wait_asynccnt 8" ::: "memory"); // slab s complete
      } else {
        asm volatile("s_wait_asynccnt 0" ::: "memory");
      }
    }
    __syncthreads();                                    // slab s visible to all waves

    const float* buf = &xs[(s & 1) * R_ROWS * D_DIM];
#pragma unroll
    for (int r = 0; r < R_ROWS; ++r) {
      const int t = s * R_ROWS + r;
      float4 xv = *(const float4*)(buf + r * D_DIM + d);
      float xx[4] = {xv.x, xv.y, xv.z, xv.w};
      float oa[4], os[4];
#pragma unroll
      for (int j = 0; j < 4; ++j) {
        float u = xx[j] - ha[j];                        // d = x - ha_old
        hv[j] = om[j] * fmaf(al[j] * u, u, hv[j]);      // (1-al)*(hv + al*d*d)
        ha[j] = fmaf(al[j], u, ha[j]);                  // al*d + ha
        oa[j] = ha[j];
        os[j] = sqrtf(hv[j]);
      }
      const size_t rowbase = (size_t)(c * L + t) * (size_t)ROW2;
      v4f oav = {oa[0], oa[1], oa[2], oa[3]};
      v4f osv = {os[0], os[1], os[2], os[3]};
      __builtin_nontemporal_store(oav, (v4f*)(out + rowbase + chain));
      __builtin_nontemporal_store(osv, (v4f*)(out + rowbase + NC + chain));
    }
    __syncthreads();                                    // done reading buf before restage
  }
}

// ---------------------------------------------------------------------------
// Fallback: plain sequential scan, 4 d's per thread (used only if d_ws tiny).
// ---------------------------------------------------------------------------
__global__ void __launch_bounds__(256)
ema_naive(const float* __restrict__ X, const float* __restrict__ h_a,
          const float* __restrict__ h_s, const float* __restrict__ alpha,
          float* __restrict__ out) {
  const int idx = blockIdx.x * blockDim.x + threadIdx.x;   // 0 .. NC/4-1
  if (idx >= NC / 4) return;
  const int chain = idx * 4;
  const int d = chain & (D_DIM - 1);

  float4 av = *(const float4*)(alpha + chain);
  float4 hav = *(const float4*)(h_a + chain);
  float4 hsv = *(const float4*)(h_s + chain);
  float al[4] = {av.x, av.y, av.z, av.w};
  float om[4], ha[4], hv[4];
  float has[4] = {hav.x, hav.y, hav.z, hav.w};
  float hss[4] = {hsv.x, hsv.y, hsv.z, hsv.w};
#pragma unroll
  for (int j = 0; j < 4; ++j) {
    om[j] = 1.0f - al[j];
    ha[j] = has[j];
    hv[j] = hss[j] * hss[j];
  }
  for (int t = 0; t < T_DIM; ++t) {
    float4 xv = *(const float4*)(X + (size_t)t * D_DIM + d);
    float xx[4] = {xv.x, xv.y, xv.z, xv.w};
    float oa[4], os[4];
#pragma unroll
    for (int j = 0; j < 4; ++j) {
      float u = xx[j] - ha[j];
      hv[j] = om[j] * fmaf(al[j] * u, u, hv[j]);
      ha[j] = fmaf(al[j], u, ha[j]);
      oa[j] = ha[j];
      os[j] = sqrtf(hv[j]);
    }
    const size_t rowbase = (size_t)t * (size_t)ROW2;
    v4f oav = {oa[0], oa[1], oa[2], oa[3]};
    v4f osv = {os[0], os[1], os[2], os[3]};
    __builtin_nontemporal_store(oav, (v4f*)(out + rowbase + chain));
    __builtin_nontemporal_store(osv, (v4f*)(out + rowbase + NC + chain));
  }
  const size_t tail = (size_t)T_DIM * (size_t)ROW2;
  *(float4*)(out + tail + chain)      = make_float4(ha[0], ha[1], ha[2], ha[3]);
  *(float4*)(out + tail + NC + chain) =
      make_float4(sqrtf(hv[0]), sqrtf(hv[1]), sqrtf(hv[2]), sqrtf(hv[3]));
}

// ---------------------------------------------------------------------------
extern "C" void kernel_launch(void* const* d_in, const int* in_sizes, int n_in,
                              void* d_out, int out_size, void* d_ws, size_t ws_size,
                              hipStream_t stream) {
  (void)in_sizes; (void)n_in; (void)out_size;
  const float* X   = (const float*)d_in[0];
  const float* h_a = (const float*)d_in[1];
  const float* h_s = (const float*)d_in[2];
  const float* al  = (const float*)d_in[3];
  float* out = (float*)d_out;
  float* ws  = (float*)d_ws;

  // Pick the largest chunk count whose scratch (7 arrays of C*NC floats) fits.
  int C = 0;
  const int cand[4] = {64, 32, 16, 8};
  for (int i = 0; i < 4; ++i) {
    if (ws_size >= (size_t)7 * (size_t)cand[i] * NC * sizeof(float)) { C = cand[i]; break; }
  }
  if (C == 0) {
    ema_naive<<<(NC / 4 + 255) / 256, 256, 0, stream>>>(X, h_a, h_s, al, out);
    return;
  }
  const int L = T_DIM / C;

  dim3 blk(512), grd(C * (N_DIM / NB_N));
  ema_phase1<<<grd, blk, 0, stream>>>(X, al, ws, C, L);
  ema_phase2<<<(NC + 255) / 256, 256, 0, stream>>>(h_a, h_s, ws, out, C);
  ema_phase3<<<grd, blk, 0, stream>>>(X, al, ws, out, C, L);
}